// SphereDeformableConv2d_20246475833533
// MI455X (gfx1250) — compile-verified
//
#include <hip/hip_runtime.h>
#include <math.h>

#define PI_F 3.14159265358979323846f

constexpr int B  = 2;
constexpr int C  = 64;
constexpr int CO = 64;
constexpr int H  = 256;
constexpr int W  = 512;
constexpr int HW = H * W;
constexpr int NPIX = B * H * W;          // 262144
constexpr int OFFC = 18;
constexpr int AFRAG_V2 = 4 * 9 * 16 * 32; // ot * tap * kchunk * lane

typedef __attribute__((ext_vector_type(2))) float v2f;
typedef __attribute__((ext_vector_type(8))) float v8f;

// ---------------------------------------------------------------------------
// Kernel 1: offset-predicting conv (3x3, zero pad) + bias + clip + sphere init
// off layout: [B][18][H][W] float, in workspace
// ---------------------------------------------------------------------------
__global__ void __launch_bounds__(256)
offsets_kernel(const float* __restrict__ x, const float* __restrict__ w_off,
               const float* __restrict__ b_off, float* __restrict__ off) {
  __shared__ float wl[OFFC * C * 9];      // 41472 B
  for (int i = threadIdx.x; i < OFFC * C * 9; i += blockDim.x) wl[i] = w_off[i];
  __syncthreads();

  int gp  = blockIdx.x * blockDim.x + threadIdx.x;   // NPIX divisible by 256
  int b   = gp / HW;
  int rem = gp - b * HW;
  int y   = rem / W;
  int xc  = rem - y * W;

  float acc[OFFC];
#pragma unroll
  for (int o = 0; o < OFFC; ++o) acc[o] = b_off[o];

  const float* xb = x + (size_t)b * C * HW;
  for (int c = 0; c < C; ++c) {
    const float* xcb = xb + c * HW;
#pragma unroll
    for (int ky = 0; ky < 3; ++ky) {
      int yy = y + ky - 1;
      if (yy < 0 || yy >= H) continue;
#pragma unroll
      for (int kx = 0; kx < 3; ++kx) {
        int xx = xc + kx - 1;
        if (xx < 0 || xx >= W) continue;
        float v = xcb[yy * W + xx];
        int widx = c * 9 + ky * 3 + kx;
#pragma unroll
        for (int o = 0; o < OFFC; ++o)
          acc[o] = fmaf(v, wl[o * C * 9 + widx], acc[o]);
      }
    }
  }

  const float maxoff = (float)(W > H ? W : H) / 50.0f;

  // sphere-aware initial offsets (analytic, per pixel)
  const float dphi = PI_F / H, dtheta = 2.0f * PI_F / W;
  const float tt = tanf(dtheta), tp = tanf(dphi), sp = tp / cosf(dtheta);
  const float phi   = -(((float)y + 0.5f) / H * PI_F - 0.5f * PI_F);
  const float theta = ((float)xc + 0.5f) / W * 2.0f * PI_F - PI_F;
  const float sphi = sinf(phi), cphi = cosf(phi);
  const float kxA[3][3] = {{-tt, 0.f, tt}, {-tt, 1.f, tt}, {-tt, 0.f, tt}};
  const float kyA[3][3] = {{sp, tp, sp}, {0.f, 1.f, 0.f}, {-sp, -tp, -sp}};

  float* dst = off + (size_t)b * OFFC * HW + rem;
#pragma unroll
  for (int k = 0; k < 9; ++k) {
    int i = k / 3, j = k % 3;
    float sy = 0.f, sx = 0.f;
    if (k != 4) {
      int kr = k % 3, kc = 2 - i;
      float kxv = kxA[kr][kc], kyv = kyA[kr][kc];
      float rho = sqrtf(kxv * kxv + kyv * kyv);
      float v  = atanf(rho);
      float sv = sinf(v), cv = cosf(v);
      float nphi   = asinf(cv * sphi + kyv * sv * cphi / rho);
      float ntheta = theta + atanf(kxv * sv / (rho * cphi * cv - kyv * sphi * sv));
      float nr = (-nphi + 0.5f * PI_F) * (float)H / PI_F - 0.5f;
      float nc = (ntheta + PI_F) * (float)W / (2.0f * PI_F) - 0.5f;
      nc = fmodf(nc, (float)W); if (nc < 0.f) nc += (float)W;   // floor-mod wrap
      sy = nr - ((float)y + (float)(i - 1));
      sx = nc - ((float)xc + (float)(j - 1));
    }
    float oy = fminf(fmaxf(acc[2 * k],     -maxoff), maxoff);
    float ox = fminf(fmaxf(acc[2 * k + 1], -maxoff), maxoff);
    dst[(size_t)(2 * k) * HW]     = oy + sy;
    dst[(size_t)(2 * k + 1) * HW] = ox + sx;
  }
}

// ---------------------------------------------------------------------------
// Kernel 2: pre-swizzle w_reg into per-lane A fragments for WMMA f32 16x16x4.
// A 16x4 layout (ISA 7.12.2): lanes 0-15 hold M=lane, {K0,K1}; lanes 16-31
// hold M=lane-16, {K2,K3}. Stored as v2f[ot][tap][kchunk][lane].
// ---------------------------------------------------------------------------
__global__ void prep_wreg(const float* __restrict__ w_reg, float* __restrict__ afrag) {
  int t = blockIdx.x * blockDim.x + threadIdx.x;
  if (t >= AFRAG_V2) return;
  int lane = t & 31;
  int q    = (t >> 5) & 15;
  int k    = (t >> 9) % 9;
  int ot   = t / (9 * 16 * 32);
  int i = k / 3, j = k % 3;
  int m  = lane & 15;
  int kb = q * 4 + ((lane >> 4) ? 2 : 0);
  int o  = ot * 16 + m;
  v2f val;
  val.x = w_reg[((o * C + kb)     * 3 + i) * 3 + j];
  val.y = w_reg[((o * C + kb + 1) * 3 + i) * 3 + j];
  ((v2f*)afrag)[t] = val;
}

// ---------------------------------------------------------------------------
// Kernel 3: fused bilinear gather + WMMA GEMM.
// 8 waves / block; wave = 16-pixel tile x 64 output channels.
// ---------------------------------------------------------------------------
__device__ __forceinline__ int sidx(int c, int p) {   // skewed LDS index
  return c * 16 + ((c >> 5) << 4) + p;                // avoids half/half bank clash
}

__global__ void __launch_bounds__(256)
sphere_dcn_main(const float* __restrict__ x, const float* __restrict__ off,
                const float* __restrict__ afrag, float* __restrict__ out) {
  constexpr int SWSZ = 64 * 16 + 16;                  // 1040 floats per wave
  __shared__ float S[8][SWSZ];

  const int lane = threadIdx.x & 31;
  const int wave = threadIdx.x >> 5;
  const int p    = lane & 15;
  const int half = lane >> 4;

  const int tile = blockIdx.x * 8 + wave;
  const int gp   = tile * 16 + p;
  const int b    = gp / HW;
  const int rem  = gp - b * HW;
  const int y    = rem / W;
  const int xc   = rem - y * W;

  const float* xb   = x + ((size_t)b * C + half * 32) * HW;
  const float* offb = off + (size_t)b * OFFC * HW + rem;
  const v2f*   af   = (const v2f*)afrag;
  float*       Sw   = S[wave];

  v8f acc0 = {}, acc1 = {}, acc2 = {}, acc3 = {};

#pragma unroll 1
  for (int k = 0; k < 9; ++k) {
    const int i = k / 3, j = k % 3;
    // ---- bilinear setup for this wave's pixel (lanes l and l+16 share p) ----
    float oy = offb[(size_t)(2 * k) * HW];
    float ox = offb[(size_t)(2 * k + 1) * HW];
    float py = (float)(y + i - 1) + oy;
    float px = (float)(xc + j - 1) + ox;
    float fy = floorf(py), fx = floorf(px);
    float dy = py - fy,    dx = px - fx;
    int y0 = (int)fy, x0 = (int)fx, y1 = y0 + 1, x1 = x0 + 1;
    bool vy0 = (y0 >= 0) && (y0 < H), vy1 = (y1 >= 0) && (y1 < H);
    bool vx0 = (x0 >= 0) && (x0 < W), vx1 = (x1 >= 0) && (x1 < W);
    float w00 = (vy0 && vx0) ? (1.f - dy) * (1.f - dx) : 0.f;
    float w01 = (vy0 && vx1) ? (1.f - dy) * dx         : 0.f;
    float w10 = (vy1 && vx0) ? dy * (1.f - dx)         : 0.f;
    float w11 = (vy1 && vx1) ? dy * dx                 : 0.f;
    int cy0 = min(max(y0, 0), H - 1), cy1 = min(max(y1, 0), H - 1);
    int cx0 = min(max(x0, 0), W - 1), cx1 = min(max(x1, 0), W - 1);
    int i00 = cy0 * W + cx0, i01 = cy0 * W + cx1;
    int i10 = cy1 * W + cx0, i11 = cy1 * W + cx1;

    // ---- gather 32 channels per lane into skewed per-wave LDS tile ----
#pragma unroll 4
    for (int c = 0; c < 32; ++c) {
      const float* base = xb + (size_t)c * HW;
      float s = w00 * base[i00] + w01 * base[i01] + w10 * base[i10] + w11 * base[i11];
      Sw[sidx(half * 32 + c, p)] = s;
    }
    __syncthreads();

    // ---- GEMM: out[o,p] += Wk[o,c] * S[c,p] via WMMA f32 16x16x4 ----
#pragma unroll 2
    for (int q = 0; q < 16; ++q) {
      int kb = q * 4 + half * 2;               // B 4x16: VGPR0 = K0/K2, VGPR1 = K1/K3
      v2f bf;
      bf.x = Sw[sidx(kb, p)];
      bf.y = Sw[sidx(kb + 1, p)];
      const v2f* afk = af + ((size_t)k * 16 + q) * 32 + lane;
      v2f a0 = afk[0];
      v2f a1 = afk[1 * 9 * 16 * 32];
      v2f a2 = afk[2 * 9 * 16 * 32];
      v2f a3 = afk[3 * 9 * 16 * 32];
      acc0 = __builtin_amdgcn_wmma_f32_16x16x4_f32(false, a0, false, bf, (short)0, acc0, false, false);
      acc1 = __builtin_amdgcn_wmma_f32_16x16x4_f32(false, a1, false, bf, (short)0, acc1, false, false);
      acc2 = __builtin_amdgcn_wmma_f32_16x16x4_f32(false, a2, false, bf, (short)0, acc2, false, false);
      acc3 = __builtin_amdgcn_wmma_f32_16x16x4_f32(false, a3, false, bf, (short)0, acc3, false, false);
    }
    __syncthreads();
  }

  // ---- store D: VGPR r holds rows r (lanes 0-15) and r+8 (lanes 16-31) ----
  const size_t ob = (size_t)b * CO * HW + rem;
#pragma unroll
  for (int r = 0; r < 8; ++r) {
    int orow = r + half * 8;
    out[ob + (size_t)(orow)      * HW] = acc0[r];
    out[ob + (size_t)(orow + 16) * HW] = acc1[r];
    out[ob + (size_t)(orow + 32) * HW] = acc2[r];
    out[ob + (size_t)(orow + 48) * HW] = acc3[r];
  }
}

// ---------------------------------------------------------------------------
extern "C" void kernel_launch(void* const* d_in, const int* in_sizes, int n_in,
                              void* d_out, int out_size, void* d_ws, size_t ws_size,
                              hipStream_t stream) {
  const float* x     = (const float*)d_in[0];
  const float* w_off = (const float*)d_in[1];
  const float* b_off = (const float*)d_in[2];
  const float* w_reg = (const float*)d_in[3];
  float* out = (float*)d_out;

  float* ws    = (float*)d_ws;
  float* off   = ws;                               // 18 * NPIX floats (~18.9 MB)
  float* afrag = ws + (size_t)OFFC * NPIX;         // 8-byte aligned, 144 KB

  offsets_kernel<<<NPIX / 256, 256, 0, stream>>>(x, w_off, b_off, off);
  prep_wreg<<<(AFRAG_V2 + 255) / 256, 256, 0, stream>>>(w_reg, afrag);
  sphere_dcn_main<<<NPIX / (16 * 8), 256, 0, stream>>>(x, off, afrag, out);
}